// Attention_73985106641132
// MI455X (gfx1250) — compile-verified
//
#include <hip/hip_runtime.h>
#include <math.h>
#include <stdint.h>

// ---------------------------------------------------------------------------
// Problem constants (from reference): B=2, S=2048, NH=16, D=64, H=1024
// ---------------------------------------------------------------------------
#define SEQ   2048
#define NHEAD 16
#define HDIM  64
#define HID   1024           // NHEAD*HDIM
#define NB    2
#define MROWS 4096           // NB*SEQ
#define N3    3072           // 3*HID

typedef float v2f __attribute__((ext_vector_type(2)));
typedef float v8f __attribute__((ext_vector_type(8)));

// CDNA5 native fp32 WMMA: D(16x16,f32) = A(16x4,f32) x B(4x16,f32) + C
static __device__ __forceinline__ v8f wmma_f32(v2f a, v2f b, v8f c) {
    return __builtin_amdgcn_wmma_f32_16x16x4_f32(false, a, false, b, (short)0, c,
                                                 false, false);
}

// CDNA5 async global->LDS copy (ASYNCcnt-tracked, no VGPR data movement).
// First operand is the LDS byte address (low 32 bits of the generic pointer
// are the workgroup-relative LDS offset), second is the 64-bit global address.
static __device__ __forceinline__ void async_ld16(float* lds, const float* g) {
    asm volatile("global_load_async_to_lds_b128 %0, %1, off"
                 :: "v"((unsigned)(unsigned long long)lds),
                    "v"((unsigned long long)g)
                 : "memory");
}
#define S_WAIT_ASYNCCNT(n) asm volatile("s_wait_asynccnt " #n ::: "memory")

// ---------------------------------------------------------------------------
// Workspace layouts (chosen by kernel 1 so flash tiles are contiguous):
//   q    : [bh][s][d]                      row-major
//   kT   : [bh][d/2][s][d&1]               transposed + d-pair interleaved
//   vI   : [bh][s/2][d][s&1]               s-pair interleaved
//   attn : [b][s][HID]                     row-major
// ---------------------------------------------------------------------------

// ---------------------------------------------------------------------------
// Kernel 1: QKV projection.  x[4096,1024] @ w[1024,3072], scatter to q/kT/vI.
// Block tile 64(M)x64(N), 8 waves of 16x32 each, K chunked by 16.
// ---------------------------------------------------------------------------
__global__ __launch_bounds__(256) void qkv_gemm_kernel(
    const float* __restrict__ x, const float* __restrict__ w,
    float* __restrict__ qo, float* __restrict__ ko, float* __restrict__ vo)
{
    __shared__ float As[64][18];      // [m][k], even pad -> aligned b64 frags
    __shared__ float Bp[8][64][2];    // [k/2][n][k&1] pair-interleaved

    const int tid  = threadIdx.x;
    const int lane = tid & 31;
    const int wv   = tid >> 5;
    const int l16  = lane & 15;
    const int khx  = lane >> 4;
    const int wm   = wv & 3;
    const int wn   = wv >> 2;
    const int m0   = blockIdx.y * 64;
    const int n0   = blockIdx.x * 64;

    v8f c0 = {}; v8f c1 = {};
    const int ar = tid >> 2, ac = (tid & 3) * 4;
    const int br = tid >> 4, bc = (tid & 15) * 4;

    for (int k0 = 0; k0 < HID; k0 += 16) {
        float4 av = *(const float4*)(x + (size_t)(m0 + ar) * HID + k0 + ac);
        As[ar][ac + 0] = av.x; As[ar][ac + 1] = av.y;
        As[ar][ac + 2] = av.z; As[ar][ac + 3] = av.w;
        float4 bv = *(const float4*)(w + (size_t)(k0 + br) * N3 + n0 + bc);
        const int bpr = br >> 1, be = br & 1;
        Bp[bpr][bc + 0][be] = bv.x; Bp[bpr][bc + 1][be] = bv.y;
        Bp[bpr][bc + 2][be] = bv.z; Bp[bpr][bc + 3][be] = bv.w;
        __syncthreads();
        #pragma unroll
        for (int kc = 0; kc < 4; ++kc) {
            v2f a  = *(const v2f*)&As[16 * wm + l16][4 * kc + 2 * khx];
            v2f b0 = *(const v2f*)&Bp[2 * kc + khx][32 * wn + l16][0];
            v2f b1 = *(const v2f*)&Bp[2 * kc + khx][32 * wn + 16 + l16][0];
            c0 = wmma_f32(a, b0, c0);
            c1 = wmma_f32(a, b1, c1);
        }
        __syncthreads();
    }

    #pragma unroll
    for (int i = 0; i < 8; ++i) {
        const int mrow = m0 + 16 * wm + i + 8 * khx;
        const int b    = mrow >> 11;
        const int s    = mrow & (SEQ - 1);
        #pragma unroll
        for (int t = 0; t < 2; ++t) {
            const int n     = n0 + 32 * wn + 16 * t + l16;
            const float val = (t == 0) ? c0[i] : c1[i];
            const int g   = n >> 10;
            const int rem = n & (HID - 1);
            const int h   = rem >> 6;
            const int d   = rem & (HDIM - 1);
            const int bh  = b * NHEAD + h;
            if (g == 0) {
                qo[((size_t)bh * SEQ + s) * HDIM + d] = val;
            } else if (g == 1) {
                ko[((size_t)(bh * 32 + (d >> 1)) * SEQ + s) * 2 + (d & 1)] = val;
            } else {
                vo[((size_t)(bh * 1024 + (s >> 1)) * HDIM + d) * 2 + (s & 1)] = val;
            }
        }
    }
}

// ---------------------------------------------------------------------------
// Kernel 2: RoPE in place on q (row-major) and kT (transposed-interleaved).
// pair (d, d+32):  out0 = x0*cos - x1*sin ; out1 = x1*cos + x0*sin
// angle = s * 10000^(-d/32)
// ---------------------------------------------------------------------------
__global__ __launch_bounds__(256) void rope_kernel(float* __restrict__ q,
                                                   float* __restrict__ kT)
{
    const int idx = blockIdx.x * 256 + threadIdx.x;   // [2][32][2048][32]
    const int d   = idx & 31;
    const int s   = (idx >> 5) & (SEQ - 1);
    const int bh  = (idx >> 16) & 31;
    const int arr = idx >> 21;

    const float inv = __powf(10000.0f, -(float)d * (1.0f / 32.0f));
    float sn, cs;
    __sincosf((float)s * inv, &sn, &cs);

    if (arr == 0) {
        float* p = q + ((size_t)bh * SEQ + s) * HDIM;
        const float x0 = p[d], x1 = p[d + 32];
        p[d]      = x0 * cs - x1 * sn;
        p[d + 32] = x1 * cs + x0 * sn;
    } else {
        float* p = kT + ((size_t)(bh * 32 + (d >> 1)) * SEQ + s) * 2 + (d & 1);
        const float x0 = p[0], x1 = p[65536];   // +16 d-pairs * 2048 s * 2
        p[0]     = x0 * cs - x1 * sn;
        p[65536] = x1 * cs + x0 * sn;
    }
}

// ---------------------------------------------------------------------------
// Kernel 3: causal flash attention.  K/V tiles are 16KB (K: 32 strided 512B rows, V: contiguous) in
// the interleaved layouts -> double-buffered async copies to LDS.
// ---------------------------------------------------------------------------
static __device__ __forceinline__ void stage_kv(const float* kT, const float* vI,
                                                int bh, int kb,
                                                float* kbuf, float* vbuf, int tid)
{
    // K tile: 32 d-pair rows x (64 krow x 2) floats   (4 async instr / wave)
    #pragma unroll
    for (int j = 0; j < 4; ++j) {
        const int qq = tid + 256 * j;           // 1024 x b128 chunks
        const int d2 = qq >> 5;
        const int c  = (qq & 31) * 4;
        const float* g = kT + ((size_t)(bh * 32 + d2) * SEQ + kb * 64) * 2 + c;
        async_ld16(kbuf + d2 * 128 + c, g);
    }
    // V tile: one contiguous 4096-float block
    const float* vg = vI + (size_t)bh * SEQ * HDIM + kb * 4096;
    #pragma unroll
    for (int j = 0; j < 4; ++j) {
        const int qq = tid + 256 * j;
        async_ld16(vbuf + qq * 4, vg + qq * 4);
    }
}

__global__ __launch_bounds__(256) void flash_attn_kernel(
    const float* __restrict__ qg, const float* __restrict__ kT,
    const float* __restrict__ vI, float* __restrict__ attn)
{
    __shared__ float Kb[2][4096];     // [d/2][krow][d&1]
    __shared__ float Vb[2][4096];     // [krow/2][d][krow&1]
    __shared__ float Ps[8][16 * 66];  // per-wave P staging, even pad

    const int tid  = threadIdx.x;
    const int lane = tid & 31;
    const int wv   = tid >> 5;
    const int l16  = lane & 15;
    const int khx  = lane >> 4;
    const int bh   = blockIdx.y;
    const int b    = bh >> 4;
    const int h    = bh & 15;
    const int qw   = blockIdx.x * 128 + wv * 16;

    const float* qh = qg + (size_t)bh * SEQ * HDIM;
    const float scale = 0.125f;

    v2f aq[16];
    #pragma unroll
    for (int kc = 0; kc < 16; ++kc) {
        const float* p = qh + (size_t)(qw + l16) * HDIM + 4 * kc + 2 * khx;
        aq[kc].x = p[0] * scale;
        aq[kc].y = p[1] * scale;
    }

    float mrow[8], lrow[8];
    v8f o0 = {}, o1 = {}, o2 = {}, o3 = {};
    #pragma unroll
    for (int i = 0; i < 8; ++i) { mrow[i] = -INFINITY; lrow[i] = 0.0f; }

    const int nkb = blockIdx.x * 2 + 2;
    stage_kv(kT, vI, bh, 0, Kb[0], Vb[0], tid);

    for (int kb = 0; kb < nkb; ++kb) {
        const int cur = kb & 1;
        if (kb + 1 < nkb) {
            stage_kv(kT, vI, bh, kb + 1, Kb[cur ^ 1], Vb[cur ^ 1], tid);
            S_WAIT_ASYNCCNT(8);       // in-order: current tile's 8 are done
        } else {
            S_WAIT_ASYNCCNT(0);
        }
        __syncthreads();

        if (kb * 64 <= qw + 15) {
            const float* kl = Kb[cur];
            const float* vl = Vb[cur];

            // scores 16x64 = 4 WMMA C-tiles, K-dim = HDIM
            v8f s0 = {}, s1 = {}, s2 = {}, s3 = {};
            #pragma unroll
            for (int kc = 0; kc < 16; ++kc) {
                const int r = (2 * kc + khx) * 128;
                v2f b0 = *(const v2f*)(kl + r + ( 0 + l16) * 2);
                v2f b1 = *(const v2f*)(kl + r + (16 + l16) * 2);
                v2f b2 = *(const v2f*)(kl + r + (32 + l16) * 2);
                v2f b3 = *(const v2f*)(kl + r + (48 + l16) * 2);
                s0 = wmma_f32(aq[kc], b0, s0);
                s1 = wmma_f32(aq[kc], b1, s1);
                s2 = wmma_f32(aq[kc], b2, s2);
                s3 = wmma_f32(aq[kc], b3, s3);
            }

            if (kb * 64 + 63 > qw) {              // diagonal block: causal mask
                const int colb = kb * 64 + l16;
                #pragma unroll
                for (int i = 0; i < 8; ++i) {
                    const int row = qw + i + 8 * khx;
                    if (colb      > row) s0[i] = -INFINITY;
                    if (colb + 16 > row) s1[i] = -INFINITY;
                    if (colb + 32 > row) s2[i] = -INFINITY;
                    if (colb + 48 > row) s3[i] = -INFINITY;
                }
            }

            // online softmax over 16-lane row groups
            #pragma unroll
            for (int i = 0; i < 8; ++i) {
                float mx = fmaxf(fmaxf(s0[i], s1[i]), fmaxf(s2[i], s3[i]));
                #pragma unroll
                for (int off = 1; off < 16; off <<= 1)
                    mx = fmaxf(mx, __shfl_xor(mx, off, 32));
                const float mnew = fmaxf(mrow[i], mx);
                const float corr = __expf(mrow[i] - mnew);
                mrow[i] = mnew;
                s0[i] = __expf(s0[i] - mnew);
                s1[i] = __expf(s1[i] - mnew);
                s2[i] = __expf(s2[i] - mnew);
                s3[i] = __expf(s3[i] - mnew);
                float sm = s0[i] + s1[i] + s2[i] + s3[i];
                #pragma unroll
                for (int off = 1; off < 16; off <<= 1)
                    sm += __shfl_xor(sm, off, 32);
                lrow[i] = lrow[i] * corr + sm;
                o0[i] *= corr; o1[i] *= corr; o2[i] *= corr; o3[i] *= corr;
                const int prow = (i + 8 * khx) * 66;
                Ps[wv][prow +  0 + l16] = s0[i];
                Ps[wv][prow + 16 + l16] = s1[i];
                Ps[wv][prow + 32 + l16] = s2[i];
                Ps[wv][prow + 48 + l16] = s3[i];
            }

            // O += P @ V   (P re-read from LDS in A-layout; V pair-interleaved)
            #pragma unroll
            for (int kc = 0; kc < 16; ++kc) {
                v2f ap = *(const v2f*)&Ps[wv][l16 * 66 + 4 * kc + 2 * khx];
                const int r = (2 * kc + khx) * 128;
                v2f b0 = *(const v2f*)(vl + r + ( 0 + l16) * 2);
                v2f b1 = *(const v2f*)(vl + r + (16 + l16) * 2);
                v2f b2 = *(const v2f*)(vl + r + (32 + l16) * 2);
                v2f b3 = *(const v2f*)(vl + r + (48 + l16) * 2);
                o0 = wmma_f32(ap, b0, o0);
                o1 = wmma_f32(ap, b1, o1);
                o2 = wmma_f32(ap, b2, o2);
                o3 = wmma_f32(ap, b3, o3);
            }
        }
        __syncthreads();
    }

    #pragma unroll
    for (int i = 0; i < 8; ++i) {
        const int srow = qw + i + 8 * khx;
        const float inv = 1.0f / lrow[i];
        float* dst = attn + (size_t)(b * SEQ + srow) * HID + h * HDIM;
        dst[ 0 + l16] = o0[i] * inv;
        dst[16 + l16] = o1[i] * inv;
        dst[32 + l16] = o2[i] * inv;
        dst[48 + l16] = o3[i] * inv;
    }
}

// ---------------------------------------------------------------------------
// Kernel 4: output projection attn[4096,1024] @ w_out[1024,1024] -> out
// ---------------------------------------------------------------------------
__global__ __launch_bounds__(256) void out_gemm_kernel(
    const float* __restrict__ a, const float* __restrict__ w,
    float* __restrict__ out)
{
    __shared__ float As[64][18];
    __shared__ float Bp[8][64][2];

    const int tid  = threadIdx.x;
    const int lane = tid & 31;
    const int wv   = tid >> 5;
    const int l16  = lane & 15;
    const int khx  = lane >> 4;
    const int wm   = wv & 3;
    const int wn   = wv >> 2;
    const int m0   = blockIdx.y * 64;
    const int n0   = blockIdx.x * 64;

    v8f c0 = {}; v8f c1 = {};
    const int ar = tid >> 2, ac = (tid & 3) * 4;
    const int br = tid >> 4, bc = (tid & 15) * 4;

    for (int k0 = 0; k0 < HID; k0 += 16) {
        float4 av = *(const float4*)(a + (size_t)(m0 + ar) * HID + k0 + ac);
        As[ar][ac + 0] = av.x; As[ar][ac + 1] = av.y;
        As[ar][ac + 2] = av.z; As[ar][ac + 3] = av.w;
        float4 bv = *(const float4*)(w + (size_t)(k0 + br) * HID + n0 + bc);
        const int bpr = br >> 1, be = br & 1;
        Bp[bpr][bc + 0][be] = bv.x; Bp[bpr][bc + 1][be] = bv.y;
        Bp[bpr][bc + 2][be] = bv.z; Bp[bpr][bc + 3][be] = bv.w;
        __syncthreads();
        #pragma unroll
        for (int kc = 0; kc < 4; ++kc) {
            v2f av2 = *(const v2f*)&As[16 * wm + l16][4 * kc + 2 * khx];
            v2f b0  = *(const v2f*)&Bp[2 * kc + khx][32 * wn + l16][0];
            v2f b1  = *(const v2f*)&Bp[2 * kc + khx][32 * wn + 16 + l16][0];
            c0 = wmma_f32(av2, b0, c0);
            c1 = wmma_f32(av2, b1, c1);
        }
        __syncthreads();
    }

    #pragma unroll
    for (int i = 0; i < 8; ++i) {
        const int mrow = m0 + 16 * wm + i + 8 * khx;
        out[(size_t)mrow * HID + n0 + 32 * wn +  0 + l16] = c0[i];
        out[(size_t)mrow * HID + n0 + 32 * wn + 16 + l16] = c1[i];
    }
}

// ---------------------------------------------------------------------------
extern "C" void kernel_launch(void* const* d_in, const int* in_sizes, int n_in,
                              void* d_out, int out_size, void* d_ws, size_t ws_size,
                              hipStream_t stream)
{
    const float* x     = (const float*)d_in[0];   // [2,2048,1024]
    const float* w_qkv = (const float*)d_in[1];   // [1024,3072]
    const float* w_out = (const float*)d_in[2];   // [1024,1024]
    float* out = (float*)d_out;                   // [2,2048,1024]

    float* ws   = (float*)d_ws;                   // 4 x 16MB
    float* q    = ws;
    float* kT   = ws + (size_t)4194304;
    float* vI   = ws + (size_t)8388608;
    float* attn = ws + (size_t)12582912;

    qkv_gemm_kernel<<<dim3(N3 / 64, MROWS / 64), 256, 0, stream>>>(x, w_qkv, q, kT, vI);
    rope_kernel<<<16384, 256, 0, stream>>>(q, kT);
    flash_attn_kernel<<<dim3(SEQ / 128, NB * NHEAD), 256, 0, stream>>>(q, kT, vI, attn);
    out_gemm_kernel<<<dim3(HID / 64, MROWS / 64), 256, 0, stream>>>(attn, w_out, out);
}